// baseline2d_53480932770435
// MI455X (gfx1250) — compile-verified
//
#include <hip/hip_runtime.h>

typedef __attribute__((ext_vector_type(2))) float v2f;
typedef __attribute__((ext_vector_type(8))) float v8f;

#define SEQ   2048
#define EMB   1280
#define H1    320
#define H2    80
#define H3    20
#define MOUT  2047        // conv output is 2047 x 2047
#define KINT  40          // interaction GEMM K (2*H3)
#define POOL  511         // floor((2047-4)/4)+1
#define NBLK  256         // pool-reduction blocks

// ---------------------------------------------------------------------------
// Generic f32 WMMA GEMM: C[M,N] = relu(A[M,K] @ W[N,K]^T + bias[N])
// Requirements: M % 16 == 0, K % 16 == 0 (for x4 unroll).
// NGUARD=false requires N % 16 == 0 (no per-fragment selects in inner loop).
// NGUARD=true clamps out-of-range columns to a valid row (unconditional b64
// loads) and zeroes the fragment with a data select (v_cndmask, no EXEC ops).
// One 16x16 output tile per wave; wave32; EXEC all-ones at WMMA.
// ---------------------------------------------------------------------------
template <bool NGUARD>
__global__ void mlp_gemm_wmma(const float* __restrict__ A,
                              const float* __restrict__ W,
                              const float* __restrict__ bias,
                              float* __restrict__ C,
                              int M, int N, int K) {
  const int lane   = threadIdx.x & 31;
  const int wave   = threadIdx.x >> 5;
  const int tilesN = (N + 15) >> 4;
  const int tilesM = M >> 4;
  const int tile   = blockIdx.x * (blockDim.x >> 5) + wave;
  if (tile >= tilesM * tilesN) return;            // wave-uniform exit
  const int mt   = tile / tilesN;
  const int nt   = tile % tilesN;
  const int half = lane >> 4;                     // 0: K 0,1  1: K 2,3
  const int l16  = lane & 15;
  const int row  = (mt << 4) + l16;               // A row this lane feeds
  const int col  = (nt << 4) + l16;               // output / W row (B column)
  const bool nok = !NGUARD || (col < N);
  const int colc = NGUARD ? (nok ? col : (N - 1)) : col;  // always valid
  const float* __restrict__ Arow = A + (size_t)row * K + half * 2;
  const float* __restrict__ Wrow = W + (size_t)colc * K + half * 2;

  v8f acc = {};
  #pragma unroll 4
  for (int k = 0; k < K; k += 4) {
    v2f a = *(const v2f*)(Arow + k);              // one global_load_b64
    v2f b = *(const v2f*)(Wrow + k);              // one global_load_b64
    if (NGUARD) {
      b.x = nok ? b.x : 0.0f;                     // v_cndmask, no EXEC games
      b.y = nok ? b.y : 0.0f;
    }
    acc = __builtin_amdgcn_wmma_f32_16x16x4_f32(false, a, false, b,
                                                (short)0, acc, false, false);
  }
  if (nok) {
    const float bv    = bias[col];
    const int   rbase = (mt << 4) + half * 8;     // C layout: vgpr t -> row t+8*half
    #pragma unroll
    for (int t = 0; t < 8; ++t) {
      float v = acc[t] + bv;
      C[(size_t)(rbase + t) * N + col] = (v > 0.0f) ? v : 0.0f;
    }
  }
}

// ---------------------------------------------------------------------------
// Build A2[2048,40]: A2[i, 0:20] = x1[i,:], A2[i, 20:40] = x1[i+1,:]; row 2047 = 0
// ---------------------------------------------------------------------------
__global__ void build_A2(const float* __restrict__ x1, float* __restrict__ A2) {
  int idx = blockIdx.x * blockDim.x + threadIdx.x;
  if (idx >= SEQ * KINT) return;
  int i = idx / KINT, k = idx % KINT;
  float v = 0.0f;
  if (i < MOUT) {
    v = (k < H3) ? x1[(size_t)i * H3 + k] : x1[(size_t)(i + 1) * H3 + (k - H3)];
  }
  A2[idx] = v;
}

// ---------------------------------------------------------------------------
// Build B2[2048,40]: B2[j,k<20]  = w00_k*x2[j,k]   + w01_k*x2[j+1,k]
//                    B2[j,k>=20] = w10_kk*x2[j,kk] + w11_kk*x2[j+1,kk]
// conv_w flat [H3][2][2]; row 2047 = 0
// ---------------------------------------------------------------------------
__global__ void build_B2(const float* __restrict__ x2,
                         const float* __restrict__ cw,
                         float* __restrict__ B2) {
  int idx = blockIdx.x * blockDim.x + threadIdx.x;
  if (idx >= SEQ * KINT) return;
  int j = idx / KINT, k = idx % KINT;
  float v = 0.0f;
  if (j < MOUT) {
    if (k < H3) {
      v = cw[k * 4 + 0] * x2[(size_t)j * H3 + k] +
          cw[k * 4 + 1] * x2[(size_t)(j + 1) * H3 + k];
    } else {
      int kk = k - H3;
      v = cw[kk * 4 + 2] * x2[(size_t)j * H3 + kk] +
          cw[kk * 4 + 3] * x2[(size_t)(j + 1) * H3 + kk];
    }
  }
  B2[idx] = v;
}

// ---------------------------------------------------------------------------
// Interaction GEMM (factorized conv): mat[i,j] = A2[i,:]·B2[j,:] + conv_b
// Padded M=N=2048 (row 2047 of A2/B2 is zero); store guarded to 2047x2047.
// 128x128 tiles of 16x16, one per wave, K=40 fully unrolled (10 WMMAs).
// ---------------------------------------------------------------------------
__global__ void interact_gemm_wmma(const float* __restrict__ A2,
                                   const float* __restrict__ B2,
                                   const float* __restrict__ conv_b,
                                   float* __restrict__ mat) {
  const int lane = threadIdx.x & 31;
  const int wave = threadIdx.x >> 5;
  const int tile = blockIdx.x * (blockDim.x >> 5) + wave;   // 0 .. 16383
  const int mt   = tile >> 7;
  const int nt   = tile & 127;
  const int half = lane >> 4;
  const int l16  = lane & 15;
  const float* __restrict__ Arow = A2 + (size_t)((mt << 4) + l16) * KINT + half * 2;
  const float* __restrict__ Brow = B2 + (size_t)((nt << 4) + l16) * KINT + half * 2;

  v8f acc = {};
  #pragma unroll
  for (int k = 0; k < KINT; k += 4) {
    v2f a = *(const v2f*)(Arow + k);
    v2f b = *(const v2f*)(Brow + k);
    acc = __builtin_amdgcn_wmma_f32_16x16x4_f32(false, a, false, b,
                                                (short)0, acc, false, false);
  }
  const float cb    = conv_b[0];
  const int   col   = (nt << 4) + l16;
  const int   rbase = (mt << 4) + half * 8;
  if (col < MOUT) {
    #pragma unroll
    for (int t = 0; t < 8; ++t) {
      const int r = rbase + t;
      if (r < MOUT) mat[(size_t)r * MOUT + col] = acc[t] + cb;
    }
  }
}

// ---------------------------------------------------------------------------
// Stage 1 reduction: 4x4 avg-pool (stride 4, VALID) then per-block max.
// ---------------------------------------------------------------------------
__global__ void pool_max(const float* __restrict__ mat, float* __restrict__ bmax) {
  __shared__ float red[256];
  const int tid    = blockIdx.x * blockDim.x + threadIdx.x;
  const int stride = gridDim.x * blockDim.x;
  float m = -3.402823466e+38f;
  for (int idx = tid; idx < POOL * POOL; idx += stride) {
    const int p = idx / POOL, q = idx % POOL;
    float s = 0.0f;
    #pragma unroll
    for (int dy = 0; dy < 4; ++dy) {
      const float* rowp = mat + (size_t)(4 * p + dy) * MOUT + 4 * q;
      #pragma unroll
      for (int dx = 0; dx < 4; ++dx) s += rowp[dx];
    }
    m = fmaxf(m, s * 0.0625f);
  }
  red[threadIdx.x] = m;
  __syncthreads();
  for (int s = 128; s > 0; s >>= 1) {
    if (threadIdx.x < s)
      red[threadIdx.x] = fmaxf(red[threadIdx.x], red[threadIdx.x + s]);
    __syncthreads();
  }
  if (threadIdx.x == 0) bmax[blockIdx.x] = red[0];
}

// ---------------------------------------------------------------------------
// Stage 2: final max over NBLK partials -> sigmoid -> pred.
// ---------------------------------------------------------------------------
__global__ void final_max(const float* __restrict__ bmax, float* __restrict__ pred) {
  __shared__ float red[256];
  float m = (threadIdx.x < NBLK) ? bmax[threadIdx.x] : -3.402823466e+38f;
  red[threadIdx.x] = m;
  __syncthreads();
  for (int s = 128; s > 0; s >>= 1) {
    if (threadIdx.x < s)
      red[threadIdx.x] = fmaxf(red[threadIdx.x], red[threadIdx.x + s]);
    __syncthreads();
  }
  if (threadIdx.x == 0) pred[0] = 1.0f / (1.0f + __expf(-red[0]));
}

// ---------------------------------------------------------------------------
extern "C" void kernel_launch(void* const* d_in, const int* in_sizes, int n_in,
                              void* d_out, int out_size, void* d_ws, size_t ws_size,
                              hipStream_t stream) {
  const float* p1     = (const float*)d_in[0];
  const float* p2     = (const float*)d_in[1];
  const float* w1     = (const float*)d_in[2];
  const float* b1     = (const float*)d_in[3];
  const float* w2     = (const float*)d_in[4];
  const float* b2     = (const float*)d_in[5];
  const float* w3     = (const float*)d_in[6];
  const float* b3     = (const float*)d_in[7];
  const float* conv_w = (const float*)d_in[8];
  const float* conv_b = (const float*)d_in[9];
  float* out = (float*)d_out;            // [0] = pred, [1..] = mat (2047*2047)
  float* mat = out + 1;

  // Workspace layout (floats)
  float* ws  = (float*)d_ws;
  float* t1a = ws;                       // 2048*320
  float* t2a = t1a + SEQ * H1;           // 2048*80
  float* x3a = t2a + SEQ * H2;           // 2048*20
  float* t1b = x3a + SEQ * H3;
  float* t2b = t1b + SEQ * H1;
  float* x3b = t2b + SEQ * H2;
  float* A2  = x3b + SEQ * H3;           // 2048*40
  float* B2  = A2 + SEQ * KINT;          // 2048*40
  float* bmx = B2 + SEQ * KINT;          // NBLK

  const dim3 blk(256);
  // MLP protein1: tiles = (M/16)*ceil(N/16); 8 waves/block
  mlp_gemm_wmma<false><<<dim3((128 * 20 + 7) / 8), blk, 0, stream>>>(p1, w1, b1, t1a, SEQ, H1, EMB);
  mlp_gemm_wmma<false><<<dim3((128 * 5  + 7) / 8), blk, 0, stream>>>(t1a, w2, b2, t2a, SEQ, H2, H1);
  mlp_gemm_wmma<true ><<<dim3((128 * 2  + 7) / 8), blk, 0, stream>>>(t2a, w3, b3, x3a, SEQ, H3, H2);
  // MLP protein2
  mlp_gemm_wmma<false><<<dim3((128 * 20 + 7) / 8), blk, 0, stream>>>(p2, w1, b1, t1b, SEQ, H1, EMB);
  mlp_gemm_wmma<false><<<dim3((128 * 5  + 7) / 8), blk, 0, stream>>>(t1b, w2, b2, t2b, SEQ, H2, H1);
  mlp_gemm_wmma<true ><<<dim3((128 * 2  + 7) / 8), blk, 0, stream>>>(t2b, w3, b3, x3b, SEQ, H3, H2);
  // Factorized conv operands
  build_A2<<<dim3((SEQ * KINT + 255) / 256), blk, 0, stream>>>(x3a, A2);
  build_B2<<<dim3((SEQ * KINT + 255) / 256), blk, 0, stream>>>(x3b, conv_w, B2);
  // Interaction GEMM -> mat (conv output), 16384 tiles / 8 waves = 2048 blocks
  interact_gemm_wmma<<<dim3(2048), blk, 0, stream>>>(A2, B2, conv_b, mat);
  // Pool + global max + sigmoid -> pred
  pool_max<<<dim3(NBLK), blk, 0, stream>>>(mat, bmx);
  final_max<<<dim3(1), blk, 0, stream>>>(bmx, out);
}